// PatchEmbedder_35003983462516
// MI455X (gfx1250) — compile-verified
//
#include <hip/hip_runtime.h>

// PatchEmbedder gather-sum for MI455X (gfx1250, wave32).
// B=8, P=1024, D=512, tables 65536x512 fp32, L = (8,7,6) ids per patch.
// Memory-bound gather: one 128-thread block (4 wave32s) per patch, float4 per
// thread. Branch-free masking (masked -> cached row 0, weight 0), patch_mask
// early-out, unconditional clustered s_loads for ids/masks (single kmcnt
// wait), global_prefetch_b8 warm-up of all 21 rows, non-temporal b128 output
// stores to keep L2 for the 384MB of gather tables.

#define D_DIM   512
#define TPB     128          // 128 * v4f == 512 floats == D
#define NPATCH  (8 * 1024)   // B * P

typedef float v4f __attribute__((ext_vector_type(4)));  // clang vector: legal
                                                        // for nontemporal builtins

template <int L>
__device__ __forceinline__ void load_meta(const int* __restrict__ ids,
                                          const int* __restrict__ msk,
                                          int patch, int* id, float* w) {
  int rid[L], rm[L];
  // Unconditional uniform loads -> compiler clusters into wide s_load_bNNN
  // with ONE s_wait_kmcnt, instead of 21 guarded+serialized s_load_b32.
#pragma unroll
  for (int l = 0; l < L; ++l) rid[l] = ids[patch * L + l];
#pragma unroll
  for (int l = 0; l < L; ++l) rm[l]  = msk[patch * L + l];
#pragma unroll
  for (int l = 0; l < L; ++l) {
    id[l] = rm[l] ? rid[l] : 0;      // branchless s_cselect; row 0 stays hot
    w[l]  = rm[l] ? 1.0f : 0.0f;
  }
}

template <int L>
__device__ __forceinline__ void prefetch_rows(const float* __restrict__ emb,
                                              const int* id, int t) {
#pragma unroll
  for (int l = 0; l < L; ++l) {
    // 128 lanes x 16B covers the whole 2KB row; global_prefetch_b8 does not
    // consume LOADcnt -> all 21 row fetches in flight before the load burst.
    __builtin_prefetch(emb + (size_t)id[l] * D_DIM + t * 4, 0, 3);
  }
}

template <int L>
__device__ __forceinline__ void accum_rows(v4f& acc,
                                           const float* __restrict__ emb,
                                           const int* id, const float* w, int t) {
#pragma unroll
  for (int l = 0; l < L; ++l) {
    const v4f v = *((const v4f*)(emb + (size_t)id[l] * D_DIM) + t);  // b128
    acc.x = fmaf(w[l], v.x, acc.x);
    acc.y = fmaf(w[l], v.y, acc.y);
    acc.z = fmaf(w[l], v.z, acc.z);
    acc.w = fmaf(w[l], v.w, acc.w);
  }
}

__global__ void __launch_bounds__(TPB)
PatchEmbedder_gather_kernel(const int*   __restrict__ ids1,
                            const int*   __restrict__ m1,
                            const float* __restrict__ e1,
                            const int*   __restrict__ ids2,
                            const int*   __restrict__ m2,
                            const float* __restrict__ e2,
                            const int*   __restrict__ ids3,
                            const int*   __restrict__ m3,
                            const float* __restrict__ e3,
                            const int*   __restrict__ pmask,
                            float*       __restrict__ out) {
  const int patch = blockIdx.x;          // b*P + p, 0..8191
  const int t     = threadIdx.x;         // 0..127, owns float4 slice of D
  v4f* outp = (v4f*)(out + (size_t)patch * D_DIM) + t;

  if (pmask[patch] == 0) {               // ~50% of patches: just write zeros
    const v4f z = {0.f, 0.f, 0.f, 0.f};
    __builtin_nontemporal_store(z, outp);
    return;
  }

  int   id1v[8]; float w1v[8];
  int   id2v[7]; float w2v[7];
  int   id3v[6]; float w3v[6];
  load_meta<8>(ids1, m1, patch, id1v, w1v);
  load_meta<7>(ids2, m2, patch, id2v, w2v);
  load_meta<6>(ids3, m3, patch, id3v, w3v);

  // Fire all 21 row prefetches before the demand-load burst.
  prefetch_rows<8>(e1, id1v, t);
  prefetch_rows<7>(e2, id2v, t);
  prefetch_rows<6>(e3, id3v, t);

  v4f acc = {0.f, 0.f, 0.f, 0.f};
  accum_rows<8>(acc, e1, id1v, w1v, t);
  accum_rows<7>(acc, e2, id2v, w2v, t);
  accum_rows<6>(acc, e3, id3v, w3v, t);

  __builtin_nontemporal_store(acc, outp);  // write-once output, keep L2 for tables
}

extern "C" void kernel_launch(void* const* d_in, const int* in_sizes, int n_in,
                              void* d_out, int out_size, void* d_ws, size_t ws_size,
                              hipStream_t stream) {
  (void)in_sizes; (void)n_in; (void)out_size; (void)d_ws; (void)ws_size;
  const int*   ids1 = (const int*)  d_in[0];
  const int*   m1   = (const int*)  d_in[1];
  const float* e1   = (const float*)d_in[2];
  const int*   ids2 = (const int*)  d_in[3];
  const int*   m2   = (const int*)  d_in[4];
  const float* e2   = (const float*)d_in[5];
  const int*   ids3 = (const int*)  d_in[6];
  const int*   m3   = (const int*)  d_in[7];
  const float* e3   = (const float*)d_in[8];
  const int*   pm   = (const int*)  d_in[9];
  float* out = (float*)d_out;

  hipLaunchKernelGGL(PatchEmbedder_gather_kernel,
                     dim3(NPATCH), dim3(TPB), 0, stream,
                     ids1, m1, e1, ids2, m2, e2, ids3, m3, e3, pm, out);
}